// ApproximatePVLFM_343597384466
// MI455X (gfx1250) — compile-verified
//
#include <hip/hip_runtime.h>
#include <hip/hip_bf16.h>

// ---------------------------------------------------------------------------
// ApproximatePVLFM on MI455X (gfx1250, wave32)
// S=512 samples, O=64 outputs, T=2048 timesteps.
// K0: hoist per-(o,step) RK4 coefficients (g0,gm,g1,a) + dt terms (2MB table).
// K1 (x8 chunks of 256 steps): per-trajectory RK4 scan; f windows DMA'd to a
//   DOUBLE-BUFFERED LDS tile with GLOBAL_LOAD_ASYNC_TO_LDS_B128 (ASYNCcnt),
//   overlapping DMA of window m+1 with compute of window m; h written u-layout
//   into a 32MB staging buffer that stays resident in the 192MB L2.
//   Steps >= 1024 only ever read f[...,2047] (index clipping) -> register.
// K2 (x8): coalesced reduction over samples; cross-sample mean via
//   V_WMMA_F32_16X16X4_F32 ones-matmul; unbiased noisy variance via VALU
//   dual-issue FMAs + wave32 shfl_xor(16) combine.
// HBM roofline: f(268MB) + u(268MB) + out(1MB) ~= 537MB -> ~23us @ 23.3TB/s.
// ---------------------------------------------------------------------------

#define S_DIM 512
#define O_DIM 64
#define T_DIM 2048
#define SO    (S_DIM * O_DIM)          // 32768
#define NSTEP (T_DIM - 1)              // 2047
#define TCH   256                      // timesteps per chunk
#define NCH   (T_DIM / TCH)            // 8

typedef float v2f __attribute__((ext_vector_type(2)));
typedef float v8f __attribute__((ext_vector_type(8)));

#if defined(__AMDGCN__) && __has_builtin(__builtin_amdgcn_wmma_f32_16x16x4_f32)
#define HAVE_WMMA_F32X4 1
#else
#define HAVE_WMMA_F32X4 0
#endif

// ---------------- K0: coefficient hoisting --------------------------------
__global__ __launch_bounds__(256) void k0_coef(
    const float* __restrict__ t, const float* __restrict__ raw_a,
    const float* __restrict__ raw_b, const float* __restrict__ raw_c,
    const float* __restrict__ raw_n,
    float4* __restrict__ coefA,   // [NSTEP][O]: (g0, gm, g1, a)
    float4* __restrict__ coefB,   // [NSTEP]   : (dt, dt/2, dt/6, 0)
    float* __restrict__ nr)       // [O]       : softplus(raw_noise)
{
  int idx = blockIdx.x * 256 + threadIdx.x;
  if (idx >= NSTEP * O_DIM) return;
  int n = idx >> 6, o = idx & 63;

  float sa = 1.0f / (1.0f + expf(-raw_a[o]));
  float sb = 1.0f / (1.0f + expf(-raw_b[o]));
  float sc = 1.0f / (1.0f + expf(-raw_c[o]));
  float a = 1e-4f + (1.0f - 1e-4f) * sa;
  float b = 1e-3f + (1.0f - 1e-3f) * sb;
  float c = 1e-3f + (1.0f - 1e-3f) * sc;

  float t0 = t[n], t1 = t[n + 1];
  float dt = t1 - t0;
  float tm = t0 + 0.5f * dt;
  const float PI = 3.14159274101257324f;   // float32(np.pi)
  float g0 = b * sinf(c * t0 * PI);
  float gm = b * sinf(c * tm * PI);
  float g1 = b * sinf(c * t1 * PI);

  coefA[idx] = make_float4(g0, gm, g1, a);
  if (o == 0) coefB[n] = make_float4(dt, 0.5f * dt, dt / 6.0f, 0.0f);
  if (n == 0) {
    float x = raw_n[o];                    // stable softplus
    nr[o] = fmaxf(x, 0.0f) + log1pf(expf(-fabsf(x)));
  }
}

// ---------------- K1: chunked RK4 scan ------------------------------------
__device__ __forceinline__ float rk4step(float h, float f1, float f2, float f34,
                                         float4 cA, float4 cB) {
  float g0 = cA.x, gm = cA.y, g1 = cA.z, a = cA.w;
  float dt = cB.x, hd = cB.y, d6 = cB.z;
  float k1 = fmaf(g0, h, -a * f1);
  float k2 = fmaf(gm, fmaf(hd, k1, h), -a * f2);
  float k3 = fmaf(gm, fmaf(hd, k2, h), -a * f34);
  float k4 = fmaf(g1, fmaf(dt, k3, h), -a * f34);
  return fmaf(d6, k1 + 2.0f * k2 + 2.0f * k3 + k4, h);
}

#define TILE_W 36        // f idx window [2*m0, 2*m0+35]; 144B row, 16B aligned

// Issue one window's 2304 16B granules as async DMA into LDS buffer `ldsbase`.
// Granules assigned row-major (9 per row) so adjacent lanes emit contiguous runs.
__device__ __forceinline__ void issue_window(const float* f, int blockBase,
                                             int fb, unsigned ldsbase, int tid) {
  #pragma unroll
  for (int i = 0; i < 9; ++i) {
    int g = tid + (i << 8);
    int r = g / 9;
    int k = g - r * 9;
    int gidx = fb + (k << 2);
    if (gidx > T_DIM - 4) gidx = T_DIM - 4;        // clamp (junk never read)
    unsigned voff = ((unsigned)(blockBase + r) * (unsigned)T_DIM +
                     (unsigned)gidx) * 4u;
    unsigned lds = ldsbase + (unsigned)(r * (TILE_W * 4) + (k << 4));
    asm volatile("global_load_async_to_lds_b128 %0, %1, %2"
                 :: "v"(lds), "v"(voff), "s"((unsigned long long)f)
                 : "memory");
  }
}

__device__ __forceinline__ void wait_async0() {
#if __has_builtin(__builtin_amdgcn_s_wait_asynccnt)
  __builtin_amdgcn_s_wait_asynccnt(0);
#else
  asm volatile("s_wait_asynccnt 0" ::: "memory");
#endif
}

__global__ __launch_bounds__(256) void k1_scan(
    const float* __restrict__ f,
    const float4* __restrict__ coefA, const float4* __restrict__ coefB,
    float* __restrict__ hstate,       // [SO] carry between chunks
    float* __restrict__ hchunk,       // [TCH][SO] staging (L2-resident)
    int c)
{
  __shared__ __align__(16) float tile[2][256 * TILE_W];   // 2 x 36KB
  const int tid = threadIdx.x;
  const int o = tid & 63;
  const int blockBase = blockIdx.x * 256;
  const int traj = blockBase + tid;                // == s*64 + o
  const int t0glob = c * TCH;

  float h = (c == 0) ? 0.5f : hstate[traj];
  hchunk[traj] = h;                                // output t_local = 0

  if (c < 4) {
    // streaming half: steps n in [c*256, c*256+255], all n <= 1023
    const unsigned lb[2] = {(unsigned)(size_t)(&tile[0][0]),
                            (unsigned)(size_t)(&tile[1][0])};
    issue_window(f, blockBase, 2 * t0glob, lb[0], tid);   // prologue: window 0

    for (int m = 0; m < 16; ++m) {
      const int p = m & 1;
      const int n0 = t0glob + m * 16;
      const int fb = 2 * n0;
      wait_async0();                // window m resident (own loads done)
      __syncthreads();              // all waves: data ready, buf 1-p reads done
      if (m + 1 < 16)               // overlap: DMA window m+1 under compute m
        issue_window(f, blockBase, fb + 32, lb[1 - p], tid);

      const float* trow = &tile[p][tid * TILE_W];
      #pragma unroll
      for (int j = 0; j < 16; ++j) {
        const int n = n0 + j;
        float4 cA = coefA[(n << 6) | o];
        float4 cB = coefB[n];
        float f1, f2, f34;
        if (n == 0) { f1 = trow[0]; f2 = trow[1]; f34 = trow[2]; }
        else {
          int l1 = 2 * n + 1 - fb;                 // in [1,31]
          int l34 = (2 * n + 2 > T_DIM - 1 ? T_DIM - 1 : 2 * n + 2) - fb;
          f1 = trow[l1]; f2 = f1; f34 = trow[l34];
        }
        h = rk4step(h, f1, f2, f34, cA, cB);
        int tl = m * 16 + j + 1;
        if (tl < TCH) hchunk[tl * SO + traj] = h;  // last step -> carry only
      }
    }
  } else {
    // tail half: all f indices clip to T-1 -> register-held force
    const float F = f[(size_t)traj * T_DIM + (T_DIM - 1)];
    for (int j = 0; j < TCH; ++j) {
      const int n = t0glob + j;
      if (n > NSTEP - 1) break;                    // n == 2047 does not exist
      float4 cA = coefA[(n << 6) | o];
      float4 cB = coefB[n];
      h = rk4step(h, F, F, F, cA, cB);
      if (j < TCH - 1) hchunk[(j + 1) * SO + traj] = h;
    }
  }
  hstate[traj] = h;
}

// ---------------- K2: reduce over samples ---------------------------------
// Block = 8 waves; wave handles one t (2 per block) x 16 o-columns x all 512 s.
// Lane L: column n = L&15, sample offset k = (L>>4)*2 (+1 for second VGPR).
__global__ __launch_bounds__(256) void k2_reduce(
    const float* __restrict__ hchunk, const float* __restrict__ u,
    const float* __restrict__ nr, float* __restrict__ out, int c)
{
  const int tid = threadIdx.x;
  const int w = tid >> 5, lane = tid & 31;
  const int og = w & 3, tw = w >> 2;
  const int tl = blockIdx.x * 2 + tw;              // local t in chunk
  const int t = c * TCH + tl;
  const int o0 = og << 4;
  const int col = lane & 15;
  const int khalf = (lane >> 4) << 1;              // 0 or 2

  const float* hrow = hchunk + (size_t)tl * SO;
  const float* urow = u + (size_t)t * SO;
  const float nrv = nr[o0 + col];
  const int offA = khalf * O_DIM + o0 + col;
  const int offB = offA + O_DIM;

  float sum_x = 0.0f, sum_x2 = 0.0f;
  v8f acc = {0.f, 0.f, 0.f, 0.f, 0.f, 0.f, 0.f, 0.f};
#if HAVE_WMMA_F32X4
  const v2f ones = {1.0f, 1.0f};
#endif

  #pragma unroll 4
  for (int s0 = 0; s0 < S_DIM; s0 += 4) {
    const int base = s0 * O_DIM;
    float h0 = hrow[base + offA];
    float h1 = hrow[base + offB];
    float u0 = urow[base + offA];
    float u1 = urow[base + offB];
#if HAVE_WMMA_F32X4
    v2f bm = {h0, h1};   // B 4x16 tile: 4 samples x 16 o-columns, N = lane%16
    acc = __builtin_amdgcn_wmma_f32_16x16x4_f32(
        false, ones, false, bm, (short)0, acc, false, false);
#else
    acc[0] += h0 + h1;
#endif
    float x0 = fmaf(u0, nrv, h0);
    float x1 = fmaf(u1, nrv, h1);
    sum_x += x0 + x1;
    sum_x2 = fmaf(x0, x0, sum_x2);
    sum_x2 = fmaf(x1, x1, sum_x2);
  }

  float sum_h = acc[0];                  // all M rows equal; per-column sum
#if !HAVE_WMMA_F32X4
  sum_h += __shfl_xor(sum_h, 16, 32);
#endif
  sum_x  += __shfl_xor(sum_x, 16, 32);
  sum_x2 += __shfl_xor(sum_x2, 16, 32);

  if (lane < 16) {
    float mean = sum_h * (1.0f / 512.0f);
    float mx   = sum_x * (1.0f / 512.0f);
    float var  = (sum_x2 - sum_x * mx) * (1.0f / 511.0f) + 1e-6f;
    out[t * O_DIM + o0 + col] = mean;                     // [0,t,o]
    out[T_DIM * O_DIM + t * O_DIM + o0 + col] = var;      // [1,t,o]
  }
}

// ---------------- launch ---------------------------------------------------
extern "C" void kernel_launch(void* const* d_in, const int* in_sizes, int n_in,
                              void* d_out, int out_size, void* d_ws, size_t ws_size,
                              hipStream_t stream) {
  const float* t_in   = (const float*)d_in[0];
  const float* f_in   = (const float*)d_in[1];
  const float* raw_a  = (const float*)d_in[2];
  const float* raw_b  = (const float*)d_in[3];
  const float* raw_c  = (const float*)d_in[4];
  const float* raw_n  = (const float*)d_in[5];
  const float* u_in   = (const float*)d_in[6];
  float* out = (float*)d_out;
  (void)in_sizes; (void)n_in; (void)out_size; (void)ws_size;

  // workspace carve-up (~34.2 MB total)
  char* w = (char*)d_ws;
  float4* coefA  = (float4*)(w + 0);              // 2047*64*16 = 2,096,128 B
  float4* coefB  = (float4*)(w + 2097152);        // 2047*16    =    32,752 B
  float*  nr     = (float*) (w + 2162688);        // 64*4
  float*  hstate = (float*) (w + 2166784);        // 32768*4 = 131,072 B
  float*  hchunk = (float*) (w + 2297856);        // 256*32768*4 = 33,554,432 B

  k0_coef<<<(NSTEP * O_DIM + 255) / 256, 256, 0, stream>>>(
      t_in, raw_a, raw_b, raw_c, raw_n, coefA, coefB, nr);

  for (int c = 0; c < NCH; ++c) {
    k1_scan<<<S_DIM / 4, 256, 0, stream>>>(f_in, coefA, coefB, hstate, hchunk, c);
    k2_reduce<<<TCH / 2, 256, 0, stream>>>(hchunk, u_in, nr, out, c);
  }
}